// FeatureGradFetcher_87170656240250
// MI455X (gfx1250) — compile-verified
//
#include <hip/hip_runtime.h>
#include <stdint.h>

// Problem constants (from reference)
#define B_  2
#define V_  5
#define C_  32
#define H_  256
#define W_  320
#define N_  40000
#define BV_ (B_*V_)
#define HW_ (H_*W_)

#define PROJ_FLOATS ((size_t)BV_*N_*2)        // per-view grid coords (ix,iy)
#define TFM_FLOATS  ((size_t)BV_*HW_*C_)      // channel-last feature maps

#ifndef __has_builtin
#define __has_builtin(x) 0
#endif

#if __has_builtin(__builtin_amdgcn_wmma_f32_16x16x4_f32)
#define USE_WMMA_PROJ 1
#else
#define USE_WMMA_PROJ 0
#endif

#if __has_builtin(__builtin_amdgcn_global_load_async_to_lds_b32) && \
    __has_builtin(__builtin_amdgcn_s_wait_asynccnt)
#define USE_ASYNC_LDS 1
#else
#define USE_ASYNC_LDS 0
#endif

typedef __attribute__((ext_vector_type(2))) float v2f;
typedef __attribute__((ext_vector_type(8))) float v8f;

// P = K @ [R|t]  (3x4), folded projection matrix
__device__ __forceinline__ void compute_P(const float* __restrict__ Kc,
                                          const float* __restrict__ Ec,
                                          float* __restrict__ P) {
#pragma unroll
  for (int i = 0; i < 3; ++i) {
    float k0 = Kc[i*3+0], k1 = Kc[i*3+1], k2 = Kc[i*3+2];
#pragma unroll
    for (int j = 0; j < 4; ++j)
      P[i*4+j] = k0*Ec[0*4+j] + k1*Ec[1*4+j] + k2*Ec[2*4+j];
  }
}

// perspective divide + grid_sample coordinate transform -> pixel-space (ix,iy)
__device__ __forceinline__ void uv_to_grid(float qx, float qy, float qz,
                                           float& ix, float& iy) {
  float inv = 1.0f / qz;
  float u = qx * inv, v = qy * inv;
  float gx = (u - 0.5f) * (2.0f / (float)(W_-1)) - 1.0f;
  float gy = (v - 0.5f) * (2.0f / (float)(H_-1)) - 1.0f;
  ix = ((gx + 1.0f) * (float)W_ - 1.0f) * 0.5f;
  iy = ((gy + 1.0f) * (float)H_ - 1.0f) * 0.5f;
}

// zero-padded bilinear tap; pixStride = element stride between pixels,
// laneOff = per-lane channel offset. All lanes of a wave share (ix,iy).
__device__ __forceinline__ float bilin(const float* __restrict__ base,
                                       float ix, float iy,
                                       int pixStride, int laneOff) {
  float x0f = floorf(ix), y0f = floorf(iy);
  float wx = ix - x0f, wy = iy - y0f;
  int x0 = (int)x0f, y0 = (int)y0f;
  int x1 = x0 + 1, y1 = y0 + 1;
  float w00 = (1.f-wx)*(1.f-wy), w10 = wx*(1.f-wy);
  float w01 = (1.f-wx)*wy,       w11 = wx*wy;
  bool vx0 = (unsigned)x0 < (unsigned)W_, vx1 = (unsigned)x1 < (unsigned)W_;
  float acc = 0.f;
  if ((unsigned)y0 < (unsigned)H_) {
    int r = y0 * W_;
    if (vx0) acc += w00 * base[(r + x0) * pixStride + laneOff];
    if (vx1) acc += w10 * base[(r + x1) * pixStride + laneOff];
  }
  if ((unsigned)y1 < (unsigned)H_) {
    int r = y1 * W_;
    if (vx0) acc += w01 * base[(r + x0) * pixStride + laneOff];
    if (vx1) acc += w11 * base[(r + x1) * pixStride + laneOff];
  }
  return acc;
}

// ---------------------------------------------------------------------------
// Kernel 1: projection. Each wave computes 16 points via one
// V_WMMA_F32_16X16X4_F32: D(16x16) = A(16x4=P padded) x B(4x16=[x;y;z;1]).
// Rows 0..2 of D give qx,qy,qz; lane n (0..15) owns point n's column.
// ---------------------------------------------------------------------------
__global__ void __launch_bounds__(256)
proj_kernel(const float* __restrict__ pts, const float* __restrict__ Kall,
            const float* __restrict__ Eall, float* __restrict__ proj) {
  const int bv = blockIdx.y;
  const int b  = bv / V_;
  float P[12];
  compute_P(Kall + bv*9, Eall + bv*12, P);
  const float* px = pts + (size_t)b * 3 * N_;

#if defined(__HIP_DEVICE_COMPILE__) && USE_WMMA_PROJ
  const int tid  = threadIdx.x;
  const int wave = tid >> 5;
  const int lane = tid & 31;
  const int col  = lane & 15;            // point within group / matrix row
  const bool hi  = (lane & 16) != 0;     // K-half selector
  const int n    = blockIdx.x * 128 + wave * 16 + col;
  const int nc   = n < N_ ? n : (N_ - 1);

  // A: 16x4, rows 0..2 = P, rest zero. lanes<16 hold K0/K1, lanes>=16 K2/K3.
  v2f a;
  a.x = (col < 3) ? P[col*4 + (hi ? 2 : 0)] : 0.f;
  a.y = (col < 3) ? P[col*4 + (hi ? 3 : 1)] : 0.f;
  // B: 4x16 = [x; y; z; 1] per point column.
  v2f bb;
  bb.x = hi ? px[2*N_ + nc] : px[nc];
  bb.y = hi ? 1.0f          : px[N_ + nc];
  v8f cz = {0.f,0.f,0.f,0.f,0.f,0.f,0.f,0.f};
  v8f d = __builtin_amdgcn_wmma_f32_16x16x4_f32(
      false, a, false, bb, (short)0, cz, false, false);
  if (!hi && n < N_) {
    float ix, iy;
    uv_to_grid(d[0], d[1], d[2], ix, iy);           // rows 0,1,2 = qx,qy,qz
    ((float2*)proj)[(size_t)bv * N_ + n] = make_float2(ix, iy);
  }
#else
  const int t = threadIdx.x;
  if (t < 128) {
    const int n = blockIdx.x * 128 + t;
    if (n < N_) {
      float x = px[n], y = px[N_ + n], z = px[2*N_ + n];
      float qx = P[0]*x + P[1]*y + P[2]*z  + P[3];
      float qy = P[4]*x + P[5]*y + P[6]*z  + P[7];
      float qz = P[8]*x + P[9]*y + P[10]*z + P[11];
      float ix, iy;
      uv_to_grid(qx, qy, qz, ix, iy);
      ((float2*)proj)[(size_t)bv * N_ + n] = make_float2(ix, iy);
    }
  }
#endif
}

// ---------------------------------------------------------------------------
// Kernel 2: [BV,C,H,W] -> [BV,H,W,C] transpose through a padded LDS tile.
// Reads: 128B coalesced along W per channel. Writes: 4KB contiguous per tile.
// Uses CDNA5 async global->LDS deposits (ASYNCcnt) when available.
// ---------------------------------------------------------------------------
__global__ void __launch_bounds__(256)
transpose_kernel(const float* __restrict__ fm, float* __restrict__ tfm) {
  __shared__ float tile[32 * 33];
  const int bv  = blockIdx.z;
  const int y   = blockIdx.y;
  const int x0  = blockIdx.x * 32;
  const int tid = threadIdx.x;
  const float* src = fm + (size_t)bv * C_ * HW_ + (size_t)y * W_ + x0;

#if defined(__HIP_DEVICE_COMPILE__) && USE_ASYNC_LDS
#pragma unroll
  for (int i = 0; i < 4; ++i) {
    int idx = i * 256 + tid;
    int c = idx >> 5, x = idx & 31;
    const float* gp = src + (size_t)c * HW_ + x;
    __builtin_amdgcn_global_load_async_to_lds_b32(
        (__attribute__((address_space(1))) int*)gp,
        (__attribute__((address_space(3))) int*)&tile[x * 33 + c],
        0, 0);
  }
  __builtin_amdgcn_s_wait_asynccnt(0);
  __syncthreads();
#else
#pragma unroll
  for (int i = 0; i < 4; ++i) {
    int idx = i * 256 + tid;
    int c = idx >> 5, x = idx & 31;
    tile[x * 33 + c] = src[(size_t)c * HW_ + x];
  }
  __syncthreads();
#endif

  float* dst = tfm + ((size_t)bv * HW_ + (size_t)y * W_ + x0) * C_;
#pragma unroll
  for (int i = 0; i < 4; ++i) {
    int idx = i * 256 + tid;
    int x = idx >> 5, c = idx & 31;
    dst[x * C_ + c] = tile[x * 33 + c];
  }
}

// ---------------------------------------------------------------------------
// Kernel 3: sampling. lane = channel (C==32==wave32). Every corner fetch is
// one contiguous 128B line from the channel-last map (L2-resident). Results
// are transposed through padded LDS so final stores are coalesced along N,
// written non-temporally (streamed, no reuse).
// ---------------------------------------------------------------------------
#define SWAVES 4
#define PTSW   32
__global__ void __launch_bounds__(128)
sample_kernel(const float* __restrict__ tfm, const float* __restrict__ proj,
              float* __restrict__ feat_out, float* __restrict__ grad_out) {
  __shared__ float featT[SWAVES * PTSW * 33];
  __shared__ v2f   gradT[SWAVES * PTSW * 33];
  const int bv   = blockIdx.y;
  const int tid  = threadIdx.x;
  const int wave = tid >> 5;
  const int lane = tid & 31;
  const int nb   = blockIdx.x * (SWAVES * PTSW) + wave * PTSW;
  const float*  t  = tfm + (size_t)bv * HW_ * C_;
  const float2* pr = (const float2*)proj + (size_t)bv * N_;
  const float SX = (float)W_ / (float)(W_ - 1);   // grid step in pixel space
  const float SY = (float)H_ / (float)(H_ - 1);
  float* fT = featT + wave * PTSW * 33;
  v2f*   gT = gradT + wave * PTSW * 33;

  for (int p = 0; p < PTSW; ++p) {
    int n = nb + p;
    float f = 0.f, gx = 0.f, gy = 0.f;
    if (n < N_) {
      float2 c = pr[n];
      f        = bilin(t, c.x,      c.y,      C_, lane);
      float fl = bilin(t, c.x - SX, c.y,      C_, lane);
      float fr = bilin(t, c.x + SX, c.y,      C_, lane);
      float ft = bilin(t, c.x,      c.y - SY, C_, lane);
      float fb = bilin(t, c.x,      c.y + SY, C_, lane);
      gx = 0.5f * (fr - fl);
      gy = 0.5f * (fb - ft);
    }
    fT[p * 33 + lane] = f;
    v2f g; g.x = gx; g.y = gy;
    gT[p * 33 + lane] = g;
  }
  __syncthreads();

  const int n = nb + lane;                // lane = point for coalesced writes
  if (n < N_) {
#pragma unroll 4
    for (int c = 0; c < C_; ++c) {
      size_t base = ((size_t)bv * C_ + c) * N_ + n;
      __builtin_nontemporal_store(fT[lane * 33 + c], feat_out + base);
      __builtin_nontemporal_store(gT[lane * 33 + c], (v2f*)grad_out + base);
    }
  }
}

// ---------------------------------------------------------------------------
// Fallback: fused kernel gathering from the original [BV,C,H,W] layout,
// used only if the workspace cannot hold the channel-last copy.
// ---------------------------------------------------------------------------
__global__ void __launch_bounds__(128)
fused_fallback(const float* __restrict__ fm, const float* __restrict__ pts,
               const float* __restrict__ Kall, const float* __restrict__ Eall,
               float* __restrict__ feat_out, float* __restrict__ grad_out) {
  const int bv = blockIdx.y;
  const int b  = bv / V_;
  float P[12];
  compute_P(Kall + bv*9, Eall + bv*12, P);
  const float* px = pts + (size_t)b * 3 * N_;
  const float* fb0 = fm + (size_t)bv * C_ * HW_;
  const int wave = threadIdx.x >> 5, lane = threadIdx.x & 31;
  const int laneOff = lane * HW_;
  const float SX = (float)W_ / (float)(W_ - 1);
  const float SY = (float)H_ / (float)(H_ - 1);
  for (int p = 0; p < PTSW; ++p) {
    int n = blockIdx.x * (4 * PTSW) + wave * PTSW + p;
    if (n >= N_) continue;
    float x = px[n], y = px[N_ + n], z = px[2*N_ + n];
    float qx = P[0]*x + P[1]*y + P[2]*z  + P[3];
    float qy = P[4]*x + P[5]*y + P[6]*z  + P[7];
    float qz = P[8]*x + P[9]*y + P[10]*z + P[11];
    float ix, iy;
    uv_to_grid(qx, qy, qz, ix, iy);
    float f  = bilin(fb0, ix,      iy,      1, laneOff);
    float fl = bilin(fb0, ix - SX, iy,      1, laneOff);
    float fr = bilin(fb0, ix + SX, iy,      1, laneOff);
    float ft = bilin(fb0, ix,      iy - SY, 1, laneOff);
    float fbm= bilin(fb0, ix,      iy + SY, 1, laneOff);
    size_t base = ((size_t)bv * C_ + lane) * N_ + n;
    feat_out[base] = f;
    grad_out[base * 2 + 0] = 0.5f * (fr - fl);
    grad_out[base * 2 + 1] = 0.5f * (fbm - ft);
  }
}

extern "C" void kernel_launch(void* const* d_in, const int* in_sizes, int n_in,
                              void* d_out, int out_size, void* d_ws, size_t ws_size,
                              hipStream_t stream) {
  (void)in_sizes; (void)n_in; (void)out_size;
  const float* fm   = (const float*)d_in[0];
  const float* pts  = (const float*)d_in[1];
  const float* Kall = (const float*)d_in[2];
  const float* Eall = (const float*)d_in[3];
  float* feat_out = (float*)d_out;
  float* grad_out = feat_out + (size_t)B_ * V_ * C_ * N_;

  const size_t req = (PROJ_FLOATS + TFM_FLOATS) * sizeof(float);
  if (ws_size >= req) {
    float* proj = (float*)d_ws;
    float* tfm  = proj + PROJ_FLOATS;
    dim3 g1((N_ + 127) / 128, BV_);
    proj_kernel<<<g1, 256, 0, stream>>>(pts, Kall, Eall, proj);
    dim3 g2(W_ / 32, H_, BV_);
    transpose_kernel<<<g2, 256, 0, stream>>>(fm, tfm);
    dim3 g3((N_ + (SWAVES * PTSW) - 1) / (SWAVES * PTSW), BV_);
    sample_kernel<<<g3, 128, 0, stream>>>(tfm, proj, feat_out, grad_out);
  } else {
    dim3 g((N_ + (4 * PTSW) - 1) / (4 * PTSW), BV_);
    fused_fallback<<<g, 128, 0, stream>>>(fm, pts, Kall, Eall, feat_out, grad_out);
  }
}